// TrafficNet_28853590294816
// MI455X (gfx1250) — compile-verified
//
#include <hip/hip_runtime.h>

typedef __attribute__((ext_vector_type(2))) float v2f;
typedef __attribute__((ext_vector_type(8))) float v8f;

constexpr int   kBatch = 262144;
constexpr int   kHid   = 64;
constexpr int   kSteps = 50;
constexpr float kBeta  = 0.95f;

// One wave32 owns 16 batch rows.
//
// State layout (per lane, 32 slots s = 2*c + r, c=0..15, r=0..1):
//   batch  = wave*16 + (lane & 15)
//   hidden = 4*c + 2*(lane>=16) + r
// which is exactly the B-operand striping of V_WMMA_F32_16X16X4_F32
// (VGPR r: K = r for lanes 0-15, K = r+2 for lanes 16-31; N = lane&15).
//
// A operand holds -W2 padded to 16 rows with rows 8,9 replicating rows 0,1,
// so the 16x16 f32 D tile carries cur2 for out0/out1 in VGPRs 0/1 on BOTH
// lane halves: lanes 0-15 store spk2, lanes 16-31 store mem2 (full-wave
// coalesced stores, EXEC stays all-ones around the WMMAs).
//
// Membranes are threshold-shifted (u = mem - 1) so Heaviside == (u > 0),
// and reset(t+1) == spk(t); spikes kept as {0,-1} so reset is an add and
// the sign folds into -W2.
__global__ __launch_bounds__(256) void snn_lif_wmma(
    const float* __restrict__ x,
    const float* __restrict__ W1,
    const float* __restrict__ b1,
    const float* __restrict__ W2,
    const float* __restrict__ b2,
    float* __restrict__ out)
{
  const int tid  = blockIdx.x * blockDim.x + threadIdx.x;
  const int wave = tid >> 5;
  const int lane = threadIdx.x & 31;
  const int half = lane >> 4;      // 0: K+0/1 rows & spk2 stores, 1: K+2/3 & mem2 stores
  const int l    = lane & 15;      // batch column within tile / A-matrix row
  const int b    = wave * 16 + l;

  const float x0 = x[2 * b + 0];
  const float x1 = x[2 * b + 1];

  float u[32];       // shifted mem1 (mem1 - 1)
  float curadj[32];  // cur1 - (1 - beta)   (loop-invariant)
  float spkneg[32];  // -spk1 in {0,-1}, doubles as next-step reset
  float w2a[32];     // -W2 in A-operand layout, rows {0,1,8,9} live

#pragma unroll
  for (int c = 0; c < 16; ++c) {
#pragma unroll
    for (int r = 0; r < 2; ++r) {
      const int s = 2 * c + r;
      const int h = 4 * c + 2 * half + r;
      curadj[s] = fmaf(x0, W1[2 * h + 0],
                  fmaf(x1, W1[2 * h + 1], b1[h])) - (1.0f - kBeta);
      u[s]      = -1.0f;           // mem1 = 0
      spkneg[s] = 0.0f;            // no reset at t=0
      const bool live = (l & 7) < 2;   // rows 0,1,8,9
      const int  o    = l & 1;
      w2a[s] = live ? -W2[o * kHid + h] : 0.0f;
    }
  }

  // C operand: b2 - (1-beta) broadcast on VGPR 0/1 (rows 0&8 / 1&9), rest 0.
  v8f cb = {};
  cb[0] = b2[0] - (1.0f - kBeta);
  cb[1] = b2[1] - (1.0f - kBeta);

  // Layer-2 state (duplicated across lane halves by construction).
  float v0 = -1.0f, v1 = -1.0f;    // shifted mem2
  float s20 = 0.0f, s21 = 0.0f;    // spk2 in {0,1}

  // Per-lane output cursor: spk2_rec region (half 0) / mem2_rec region (half 1).
  float* p = out + (size_t)2 * b
                 + (half ? (size_t)kSteps * kBatch * 2 : (size_t)0);

#pragma unroll 1
  for (int t = 0; t < kSteps; ++t) {
    // ---- layer 1 LIF: u' = beta*u + curadj - spk_prev ; spk = (u' > 0) ----
#pragma unroll
    for (int s = 0; s < 32; ++s) {
      u[s]      = fmaf(kBeta, u[s], curadj[s]) + spkneg[s];
      spkneg[s] = (u[s] > 0.0f) ? -1.0f : 0.0f;
    }

    // ---- cur2adj = spk1 @ W2^T + b2 - (1-beta): 16 chained f32 WMMAs ----
    v8f d = cb;
#pragma unroll
    for (int c = 0; c < 16; ++c) {
      v2f a;  a.x  = w2a[2 * c];     a.y  = w2a[2 * c + 1];
      v2f bm; bm.x = spkneg[2 * c];  bm.y = spkneg[2 * c + 1];
      d = __builtin_amdgcn_wmma_f32_16x16x4_f32(
              /*neg_a=*/false, a, /*neg_b=*/false, bm,
              /*c_mod=*/(short)0, d, /*reuse_a=*/false, /*reuse_b=*/false);
    }

    // ---- layer 2 LIF ----
    v0 = fmaf(kBeta, v0, d[0]) - s20;
    v1 = fmaf(kBeta, v1, d[1]) - s21;
    s20 = (v0 > 0.0f) ? 1.0f : 0.0f;
    s21 = (v1 > 0.0f) ? 1.0f : 0.0f;

    // half 0 lanes write spk2, half 1 lanes write mem2 (= v + 1).
    v2f outv;
    outv.x = half ? (v0 + 1.0f) : s20;
    outv.y = half ? (v1 + 1.0f) : s21;
    __builtin_nontemporal_store(outv, (v2f*)p);   // 8B/lane, full-wave coalesced
    p += (size_t)kBatch * 2;
  }
}

extern "C" void kernel_launch(void* const* d_in, const int* in_sizes, int n_in,
                              void* d_out, int out_size, void* d_ws, size_t ws_size,
                              hipStream_t stream) {
  const float* x  = (const float*)d_in[0];
  const float* W1 = (const float*)d_in[1];
  const float* b1 = (const float*)d_in[2];
  const float* W2 = (const float*)d_in[3];
  const float* b2 = (const float*)d_in[4];
  float* out = (float*)d_out;

  const int waves_total = kBatch / 16;          // 16384 waves
  const int block = 256;                        // 8 waves / block
  const int grid  = waves_total / (block / 32); // 2048 blocks

  hipLaunchKernelGGL(snn_lif_wmma, dim3(grid), dim3(block), 0, stream,
                     x, W1, b1, W2, b2, out);
}